// GlobalStateBus_18356690223153
// MI455X (gfx1250) — compile-verified
//
#include <hip/hip_runtime.h>

typedef __attribute__((ext_vector_type(2))) float v2f;
typedef __attribute__((ext_vector_type(8))) float v8f;

#define B_  128
#define DQ_ 1024
#define S_  64
#define P_  8
#define V_  512

// ---------------------------------------------------------------------------
// Phase 1: out[M,N] = act(A[M,K] @ W[K,N] + bias[N]) with fp32 WMMA 16x16x4.
// One wave (32 threads) per 16x16 output tile; EXEC is all-ones (no divergence
// before the WMMAs), as required by the ISA.
//   A layout: lane 0-15 -> M=lane, VGPR0/1 = K=k0,k0+1 ; lane 16-31 -> K=k0+2,k0+3
//   B layout: lane&15 -> N column, lane half selects K pair (mirrors A)
//   C layout: VGPR r -> row M=r (lanes 0-15) / M=r+8 (lanes 16-31), N=lane&15
// ---------------------------------------------------------------------------
__global__ __launch_bounds__(32)
void wmma_gemm_f32(const float* __restrict__ A, const float* __restrict__ W,
                   const float* __restrict__ bias, float* __restrict__ out,
                   int M, int K, int N, int act_sigmoid)
{
    const int lane = threadIdx.x;        // 0..31
    const int tn   = blockIdx.x * 16;    // N tile base
    const int tm   = blockIdx.y * 16;    // M tile base
    const int half = lane >> 4;          // 0 or 1
    const int l15  = lane & 15;
    const int arow = tm + l15;           // A row for this lane
    const int bcol = tn + l15;           // B column for this lane

    v8f c = {};
    for (int k0 = 0; k0 < K; k0 += 4) {
        const int kk = k0 + (half << 1);
        // A[arow][kk], A[arow][kk+1] are contiguous -> one b64 load
        v2f a = *(const v2f*)&A[(size_t)arow * K + kk];
        v2f b;
        b.x = W[(size_t)kk       * N + bcol];
        b.y = W[(size_t)(kk + 1) * N + bcol];
        // (neg_a, A, neg_b, B, c_mod, C, reuse_a, reuse_b)
        c = __builtin_amdgcn_wmma_f32_16x16x4_f32(false, a, false, b,
                                                  (short)0, c, false, false);
    }

    const float bb = bias[bcol];
#pragma unroll
    for (int r = 0; r < 8; ++r) {
        float v = c[r] + bb;
        if (act_sigmoid) v = 1.0f / (1.0f + expf(-v));
        out[(size_t)(tm + half * 8 + r) * N + bcol] = v;
    }
}

// ---------------------------------------------------------------------------
// Phase 2: fused EMA + scores + softmax(P) + collapse. One 256-thread block
// per (b,s) pair; each thread owns v0=2t, v1=2t+1 across all P=8 slots.
// Single streaming read of the 128 MiB buffer (the bandwidth floor).
// ---------------------------------------------------------------------------
__global__ __launch_bounds__(256)
void collapse_kernel(const float* __restrict__ buffer,
                     const float* __restrict__ gate,
                     const float* __restrict__ content,
                     const float* __restrict__ qproj,
                     float* __restrict__ out)
{
    const int bs = blockIdx.x;        // 0..B*S-1
    const int b  = bs >> 6;           // / S_
    const int s  = bs & (S_ - 1);
    const int t  = threadIdx.x;       // 0..255
    const int v0 = t * 2;

    __shared__ float s_dot[P_];
    if (t < P_) s_dot[t] = 0.0f;
    __syncthreads();

    const float g  = gate[b * S_ + s];
    const v2f   q  = *(const v2f*)&qproj[(size_t)b * V_ + v0];
    const v2f   cv = *(const v2f*)&content[(size_t)b * V_ + v0];

    const float* __restrict__ bufb = buffer + (size_t)bs * P_ * V_;

    v2f   bp[P_];
    float part[P_];
#pragma unroll
    for (int p = 0; p < P_; ++p) {
        bp[p]   = *(const v2f*)&bufb[(size_t)p * V_ + v0];
        part[p] = bp[p].x * q.x + bp[p].y * q.y;
    }

    // wave32 butterfly reduction of the 8 partial dot products
#pragma unroll
    for (int p = 0; p < P_; ++p) {
#pragma unroll
        for (int off = 16; off >= 1; off >>= 1)
            part[p] += __shfl_xor(part[p], off, 32);
    }
    if ((t & 31) == 0) {
#pragma unroll
        for (int p = 0; p < P_; ++p) atomicAdd(&s_dot[p], part[p]);
    }
    __syncthreads();

    // softmax over P with score_p = (1-g) * dot_p / sqrt(V)
    // (the g*(content.q) term is constant over p and cancels in softmax)
    const float scale = (1.0f - g) * 0.044194173824159216f;  // 1/sqrt(512)
    float sc[P_];
    float m = -3.402823466e+38f;
#pragma unroll
    for (int p = 0; p < P_; ++p) { sc[p] = s_dot[p] * scale; m = fmaxf(m, sc[p]); }
    float sum = 0.0f;
#pragma unroll
    for (int p = 0; p < P_; ++p) { sc[p] = expf(sc[p] - m); sum += sc[p]; }
    const float inv = 1.0f / sum;

    float o0 = 0.0f, o1 = 0.0f;
#pragma unroll
    for (int p = 0; p < P_; ++p) {
        const float w = sc[p] * inv;
        o0 += w * bp[p].x;
        o1 += w * bp[p].y;
    }

    v2f o;
    o.x = (1.0f - g) * o0 + g * cv.x;   // sum_p w_p = 1
    o.y = (1.0f - g) * o1 + g * cv.y;
    *(v2f*)&out[(size_t)bs * V_ + v0] = o;
}

// ---------------------------------------------------------------------------
extern "C" void kernel_launch(void* const* d_in, const int* in_sizes, int n_in,
                              void* d_out, int out_size, void* d_ws, size_t ws_size,
                              hipStream_t stream)
{
    // setup_inputs order: query, x, buffer, Wc, bc, Wg, bg, Wv, bv (all fp32)
    const float* query  = (const float*)d_in[0];
    const float* x      = (const float*)d_in[1];
    const float* buffer = (const float*)d_in[2];
    const float* Wc     = (const float*)d_in[3];
    const float* bc     = (const float*)d_in[4];
    const float* Wg     = (const float*)d_in[5];
    const float* bg     = (const float*)d_in[6];
    const float* Wv     = (const float*)d_in[7];
    const float* bv     = (const float*)d_in[8];
    float* out = (float*)d_out;

    // workspace layout (floats): gate[B*S] | content[B*V] | qproj[B*V]
    float* gate    = (float*)d_ws;
    float* content = gate + B_ * S_;
    float* qproj   = content + B_ * V_;

    // gate = sigmoid(x @ Wg + bg)   (128x64)
    wmma_gemm_f32<<<dim3(S_ / 16, B_ / 16), 32, 0, stream>>>(
        x, Wg, bg, gate, B_, DQ_, S_, 1);
    // content = x @ Wv + bv         (128x512)
    wmma_gemm_f32<<<dim3(V_ / 16, B_ / 16), 32, 0, stream>>>(
        x, Wv, bv, content, B_, DQ_, V_, 0);
    // qproj = query @ Wc + bc       (128x512)
    wmma_gemm_f32<<<dim3(V_ / 16, B_ / 16), 32, 0, stream>>>(
        query, Wc, bc, qproj, B_, DQ_, V_, 0);

    // fused collapse, one block per (b,s)
    collapse_kernel<<<B_ * S_, 256, 0, stream>>>(buffer, gate, content, qproj, out);
}